// PredictionNet_71889162600856
// MI455X (gfx1250) — compile-verified
//
#include <hip/hip_runtime.h>
#include <math.h>
#include <stdint.h>

typedef float v2f __attribute__((ext_vector_type(2)));
typedef float v8f __attribute__((ext_vector_type(8)));

#define NEXP 6

// ---- CDNA5 async global->LDS copy (ASYNCcnt-tracked), per CDNA5 ISA.
// LDS operand is the 32-bit LDS byte address (low 32 bits of the flat address).
__device__ __forceinline__ void async_ld_b128(uint32_t lds_off, const float* g) {
    asm volatile("global_load_async_to_lds_b128 %0, %1, off"
                 :: "v"(lds_off), "v"(g) : "memory");
}
__device__ __forceinline__ void wait_async0() {
    asm volatile("s_wait_asynccnt 0x0" ::: "memory");
}

// ---------------------------------------------------------------------------
// prep: h0 = concat(z, p_prev) [256,1664]; z prefix (cols 0..31) of h1cat/h2cat.
// ---------------------------------------------------------------------------
__global__ void prep_kernel(const float* __restrict__ z,
                            const float* __restrict__ p_prev,
                            float* __restrict__ h0,
                            float* __restrict__ h1cat,
                            float* __restrict__ h2cat) {
    int idx = blockIdx.x * blockDim.x + threadIdx.x;
    if (idx >= 256 * 1664) return;
    int b = idx / 1664;
    int c = idx - b * 1664;
    float v;
    if (c < 32) {
        v = z[b * 32 + c];
        h1cat[b * 544 + c] = v;
        h2cat[b * 544 + c] = v;
    } else {
        v = p_prev[b * 1632 + (c - 32)];
    }
    h0[idx] = v;
}

// ---------------------------------------------------------------------------
// Per-expert GEMM: Y[e] = H @ W[e]   (e = blockIdx.z)
//   H [256, LDH] (uses first I cols), W [6, I, O], Y [6, 256, O]
// Block: 128 thr (4 waves). Block tile 16(M) x 256(N); wave tile 16x64 via 4
// V_WMMA_F32_16X16X4_F32 accumulators. A tiles (16x32) double-buffered in LDS
// via global_load_async_to_lds_b128; B loads go saddr+imm (O compile-time).
// GUARD: clamp B columns >= O (loads stay in-bounds), predicate stores only.
// Requires I % 32 == 0.
// ---------------------------------------------------------------------------
template<int I, int O, int LDH, bool GUARD>
__global__ __launch_bounds__(128) void expert_gemm_kernel(
    const float* __restrict__ H, const float* __restrict__ W,
    float* __restrict__ Y)
{
    constexpr int CPE = I / 32;          // K-chunks for this expert
    constexpr int AP  = 36;              // padded LDS row (floats): 16B-aligned,
                                         // 36*m % 64 distinct -> conflict-free

    __shared__ float As[2][16 * AP];

    const int tid  = threadIdx.x;
    const int lane = tid & 31;
    const int wave = tid >> 5;
    const int l16  = lane & 15;          // A row / B,C,D col within 16
    const int hi   = lane >> 4;          // lane half: K-pair & M+8 half
    const int kb   = hi * 2;

    const int e     = blockIdx.z;
    const int bm0   = blockIdx.x * 16;
    const int nBase = blockIdx.y * 256 + wave * 64;

    const float* We = W + (size_t)e * I * O;
    float*       Ye = Y + (size_t)e * 256 * O;

    // A staging: thread -> (row am, 4 consecutive floats at ak) = one b128/lane
    const int am = tid >> 3;             // 0..15
    const int ak = (tid & 7) * 4;        // 0..28
    const float*   aSrc  = H + (size_t)(bm0 + am) * LDH + ak;
    const uint32_t aDst0 = (uint32_t)(uintptr_t)&As[0][am * AP + ak];
    const uint32_t aDst1 = (uint32_t)(uintptr_t)&As[1][am * AP + ak];

    // Per-lane B column offsets (clamped in GUARD mode -> loads in-bounds)
    int ntc[4];
    #pragma unroll
    for (int t = 0; t < 4; ++t) {
        int nt = nBase + t * 16 + l16;
        ntc[t] = GUARD ? (nt < O ? nt : O - 1) : nt;
    }

    v8f acc[4] = {};

    async_ld_b128(aDst0, aSrc);          // prefetch chunk 0

    for (int ci = 0; ci < CPE; ++ci) {
        wait_async0();                   // my lanes' async copy done
        __syncthreads();                 // => all waves' copies done, prev reads done
        if (ci + 1 < CPE)                // prefetch next chunk into other buffer
            async_ld_b128((ci & 1) ? aDst0 : aDst1, aSrc + (ci + 1) * 32);

        const float* as_ = &As[ci & 1][0];
        const float* wb  = We + (size_t)ci * 32 * O;   // uniform scalar base
        #pragma unroll
        for (int k4 = 0; k4 < 8; ++k4) {
            v2f a;
            a.x = as_[l16 * AP + k4 * 4 + kb];
            a.y = as_[l16 * AP + k4 * 4 + kb + 1];
            #pragma unroll
            for (int t = 0; t < 4; ++t) {
                v2f bf;                  // saddr + vgpr(ntc) + imm(k*O*4)
                bf.x = wb[(k4 * 4 + kb) * O + ntc[t]];
                bf.y = wb[(k4 * 4 + kb + 1) * O + ntc[t]];
                acc[t] = __builtin_amdgcn_wmma_f32_16x16x4_f32(
                    false, a, false, bf, (short)0, acc[t], false, false);
            }
        }
    }

    // store Y tile (C/D layout: VGPR j holds rows M=j and M=j+8)
    #pragma unroll
    for (int j = 0; j < 8; ++j) {
        const int r = bm0 + j + hi * 8;
        float* yr = Ye + (size_t)r * O;
        #pragma unroll
        for (int t = 0; t < 4; ++t) {
            int nt = nBase + t * 16 + l16;
            if (!GUARD || nt < O) yr[nt] = acc[t][j];
        }
    }
}

// ---------------------------------------------------------------------------
// blend: out[b,o] = act( sum_e coef[b,e] * (Y[e,b,o] + bias[e,o]) )
// ---------------------------------------------------------------------------
template<int O, int LDO, bool ELU>
__global__ void blend_kernel(const float* __restrict__ Y,
                             const float* __restrict__ bias,
                             const float* __restrict__ coef,
                             float* __restrict__ out)
{
    int idx = blockIdx.x * blockDim.x + threadIdx.x;
    if (idx >= 256 * O) return;
    int b = idx / O;
    int o = idx - b * O;
    float s = 0.0f;
    #pragma unroll
    for (int e = 0; e < NEXP; ++e) {
        float cf = coef[b * NEXP + e];
        s = fmaf(cf, Y[((size_t)e * 256 + b) * O + o] + bias[e * O + o], s);
    }
    if (ELU) s = (s > 0.0f) ? s : expm1f(s);   // jax.nn.elu, alpha=1
    out[(size_t)b * LDO + o] = s;
}

// ---------------------------------------------------------------------------
// Inputs (all f32): 0 p_prev[256,1632] 1 coef[256,6] 2 z[256,32]
//  3 w_l1[6,1664,512] 4 b_l1[6,512] 5 w_l2[6,544,512] 6 b_l2[6,512]
//  7 w_l3[6,544,618] 8 b_l3[6,618].  Output [256,618] f32.
// ---------------------------------------------------------------------------
extern "C" void kernel_launch(void* const* d_in, const int* in_sizes, int n_in,
                              void* d_out, int out_size, void* d_ws, size_t ws_size,
                              hipStream_t stream) {
    (void)in_sizes; (void)n_in; (void)out_size; (void)ws_size;
    const float* p_prev = (const float*)d_in[0];
    const float* coef   = (const float*)d_in[1];
    const float* z      = (const float*)d_in[2];
    const float* w1     = (const float*)d_in[3];
    const float* b1     = (const float*)d_in[4];
    const float* w2     = (const float*)d_in[5];
    const float* b2     = (const float*)d_in[6];
    const float* w3     = (const float*)d_in[7];
    const float* b3     = (const float*)d_in[8];
    float* out = (float*)d_out;

    // ws: h0[256,1664] | h1cat[256,544] | h2cat[256,544] | Y[6,256,618] ~6.6MB
    float* h0    = (float*)d_ws;
    float* h1cat = h0 + 256 * 1664;
    float* h2cat = h1cat + 256 * 544;
    float* Ybuf  = h2cat + 256 * 544;

    prep_kernel<<<(256 * 1664 + 255) / 256, 256, 0, stream>>>(z, p_prev, h0, h1cat, h2cat);

    // layer1: Y_e = h0 @ W1_e ; blend+ELU -> h1cat cols 32..543
    expert_gemm_kernel<1664, 512, 1664, false><<<dim3(16, 2, 6), 128, 0, stream>>>(h0, w1, Ybuf);
    blend_kernel<512, 544, true><<<(256 * 512 + 255) / 256, 256, 0, stream>>>(Ybuf, b1, coef, h1cat + 32);

    // layer2
    expert_gemm_kernel<544, 512, 544, false><<<dim3(16, 2, 6), 128, 0, stream>>>(h1cat, w2, Ybuf);
    blend_kernel<512, 544, true><<<(256 * 512 + 255) / 256, 256, 0, stream>>>(Ybuf, b2, coef, h2cat + 32);

    // layer3 (O=618 -> guarded tail, no activation)
    expert_gemm_kernel<544, 618, 544, true><<<dim3(16, 3, 6), 128, 0, stream>>>(h2cat, w3, Ybuf);
    blend_kernel<618, 618, false><<<(256 * 618 + 255) / 256, 256, 0, stream>>>(Ybuf, b3, coef, out);
}